// GraphLearner_64665027608710
// MI455X (gfx1250) — compile-verified
//
#include <hip/hip_runtime.h>
#include <hip/hip_bf16.h>

#define B_SZ 8
#define N_SZ 2048
#define D_SZ 512
#define H_SZ 512
#define TOPK 10
#define NEGINF (-1e20f)
#define SSTRIDE 516        // 16-row strip stride in floats; 516%64 offsets banks per row-half
#define NPASS 4
#define PASSCOLS 512
#define NCAND (NPASS * TOPK)   // 40 candidates per row

typedef __attribute__((ext_vector_type(2))) float v2f;
typedef __attribute__((ext_vector_type(8))) float v8f;

static __device__ __forceinline__ v8f wmma4(v2f a, v2f b, v8f c) {
  // V_WMMA_F32_16X16X4_F32 : D = A(16x4) x B(4x16) + C, exact fp32
  return __builtin_amdgcn_wmma_f32_16x16x4_f32(
      /*neg_a=*/false, a, /*neg_b=*/false, b,
      /*c_mod=*/(short)0, c, /*reuse_a=*/false, /*reuse_b=*/false);
}

// Intra-wave LDS write->read ordering (lane0 clears a slot, all lanes re-read).
static __device__ __forceinline__ void lds_fence_wave() {
  __builtin_amdgcn_wave_barrier();
  asm volatile("s_wait_dscnt 0" ::: "memory");
}

// ---------------------------------------------------------------------------
// Kernel 1: fc = relu(context @ W^T), [B*N, D] x [D, H] -> [B*N, H]
// One wave per 16x16 output tile; fp32 WMMA 16x16x4, two accumulator chains.
// ---------------------------------------------------------------------------
__global__ void __launch_bounds__(128) k_fc(const float* __restrict__ ctx,
                                            const float* __restrict__ W,
                                            float* __restrict__ fc) {
  const int lane = threadIdx.x & 31;
  const int wid  = threadIdx.x >> 5;
  const int rowTile = blockIdx.x;             // (B*N)/16 = 1024
  const int colTile = blockIdx.y * 4 + wid;   // H/16 = 32
  const int m    = lane & 15;
  const int koff = (lane >> 4) * 2;           // A/B fp32 layout: K pair 0..1 / 2..3

  const float* aptr = ctx + ((size_t)rowTile * 16 + m) * D_SZ + koff;
  const float* bptr = W   + ((size_t)colTile * 16 + m) * D_SZ + koff; // B[k][n]=W[n][k]

  v8f c0 = {}; v8f c1 = {};
#pragma unroll 4
  for (int k0 = 0; k0 < D_SZ; k0 += 8) {
    v2f a0 = *(const v2f*)(aptr + k0);
    v2f b0 = *(const v2f*)(bptr + k0);
    v2f a1 = *(const v2f*)(aptr + k0 + 4);
    v2f b1 = *(const v2f*)(bptr + k0 + 4);
    c0 = wmma4(a0, b0, c0);
    c1 = wmma4(a1, b1, c1);
  }
  const int rbase = (lane >> 4) * 8;          // C layout: lanes16-31 hold M = v+8
#pragma unroll
  for (int v = 0; v < 8; ++v) {
    float x = c0[v] + c1[v];
    x = x > 0.f ? x : 0.f;                    // ReLU
    fc[((size_t)rowTile * 16 + rbase + v) * H_SZ + colTile * 16 + m] = x;
  }
}

// ---------------------------------------------------------------------------
// Kernel 2: fill output with NEG_INF (float4 streaming stores)
// ---------------------------------------------------------------------------
__global__ void __launch_bounds__(256) k_fill(float4* __restrict__ out) {
  size_t i = (size_t)blockIdx.x * 256 + threadIdx.x;
  out[i] = make_float4(NEGINF, NEGINF, NEGINF, NEGINF);
}

// ---------------------------------------------------------------------------
// Kernel 3: per (batch, 16-row tile): att strip in LDS + masked top-10 scatter.
// 256 threads = 8 waves. 4 passes of 512 columns. Never writes att to HBM.
// ---------------------------------------------------------------------------
__global__ void __launch_bounds__(256) k_att(const float* __restrict__ fc,
                                             const int* __restrict__ mask,
                                             float* __restrict__ out) {
  __shared__ float strip[16 * SSTRIDE];   // ~33 KB
  __shared__ float cv[16 * NCAND];
  __shared__ int   ci[16 * NCAND];
  __shared__ int   rowm[16];

  const int bb   = blockIdx.x >> 7;       // N/16 = 128 row tiles per batch
  const int rt   = blockIdx.x & 127;
  const int row0 = rt * 16;
  const int tid  = threadIdx.x;
  const int lane = tid & 31;
  const int wid  = tid >> 5;
  const int m    = lane & 15;
  const int koff = (lane >> 4) * 2;
  const int rbase = (lane >> 4) * 8;

  if (tid < 16) rowm[tid] = mask[bb * N_SZ + row0 + tid];
  __syncthreads();

  const float* abase = fc + ((size_t)bb * N_SZ + row0 + m) * H_SZ + koff;

  for (int p = 0; p < NPASS; ++p) {
    const int passBase = p * PASSCOLS;

    // --- compute 16 x 512 strip of att = fc @ fc^T via fp32 WMMA ---
    for (int t = 0; t < 4; ++t) {
      const int colLocal = (wid * 4 + t) * 16;       // 0..496
      const int col = passBase + colLocal + m;
      const float* bptr = fc + ((size_t)bb * N_SZ + col) * H_SZ + koff; // B[k][n]=fc[col][k]
      v8f c0 = {}; v8f c1 = {};
#pragma unroll 4
      for (int k0 = 0; k0 < H_SZ; k0 += 8) {
        v2f a0 = *(const v2f*)(abase + k0);
        v2f b0 = *(const v2f*)(bptr + k0);
        v2f a1 = *(const v2f*)(abase + k0 + 4);
        v2f b1 = *(const v2f*)(bptr + k0 + 4);
        c0 = wmma4(a0, b0, c0);
        c1 = wmma4(a1, b1, c1);
      }
      const int cm = mask[bb * N_SZ + col];
#pragma unroll
      for (int v = 0; v < 8; ++v) {
        const int r = rbase + v;
        float x = c0[v] + c1[v];
        const bool ok = (cm != 0) && (rowm[r] != 0);
        strip[r * SSTRIDE + colLocal + m] = ok ? x : NEGINF;
      }
    }
    __syncthreads();

    // --- per-pass top-10 per row: iterative wave argmax (stable: lowest idx) ---
    for (int rr = 0; rr < 2; ++rr) {
      const int row = wid * 2 + rr;
      float* srow = strip + row * SSTRIDE;
      for (int sel = 0; sel < TOPK; ++sel) {
        float best = srow[lane];
        int bidx = lane;
        for (int j = 32 + lane; j < PASSCOLS; j += 32) {
          const float v = srow[j];
          if (v > best) { best = v; bidx = j; }   // strict > keeps lowest index
        }
#pragma unroll
        for (int off = 16; off; off >>= 1) {
          const float ov = __shfl_xor(best, off, 32);
          const int   oi = __shfl_xor(bidx, off, 32);
          if (ov > best || (ov == best && oi < bidx)) { best = ov; bidx = oi; }
        }
        if (lane == 0) {
          cv[row * NCAND + p * TOPK + sel] = best;
          ci[row * NCAND + p * TOPK + sel] = passBase + bidx;
          srow[bidx] = NEGINF;                    // remove winner
        }
        lds_fence_wave();
      }
    }
    __syncthreads();
  }

  // --- final: top-10 of the 40 candidates, scatter into pre-filled output ---
  for (int rr = 0; rr < 2; ++rr) {
    const int row = wid * 2 + rr;
    const size_t obase = ((size_t)bb * N_SZ + row0 + row) * (size_t)N_SZ;
    for (int sel = 0; sel < TOPK; ++sel) {
      float v1 = cv[row * NCAND + lane];
      int   cc = ci[row * NCAND + lane];
      int   pp = lane;
      if (lane < NCAND - 32) {                    // lanes 0..7 also cover 32..39
        const float v2 = cv[row * NCAND + 32 + lane];
        const int   c2 = ci[row * NCAND + 32 + lane];
        if (v2 > v1 || (v2 == v1 && c2 < cc)) { v1 = v2; cc = c2; pp = 32 + lane; }
      }
#pragma unroll
      for (int off = 16; off; off >>= 1) {
        const float ov = __shfl_xor(v1, off, 32);
        const int   oc = __shfl_xor(cc, off, 32);
        const int   op = __shfl_xor(pp, off, 32);
        if (ov > v1 || (ov == v1 && oc < cc)) { v1 = ov; cc = oc; pp = op; }
      }
      if (lane == 0) {
        out[obase + cc] = v1;                     // scatter top-k value
        cv[row * NCAND + pp] = NEGINF;            // remove from candidate pool
      }
      lds_fence_wave();
    }
  }
}

// ---------------------------------------------------------------------------
extern "C" void kernel_launch(void* const* d_in, const int* in_sizes, int n_in,
                              void* d_out, int out_size, void* d_ws, size_t ws_size,
                              hipStream_t stream) {
  (void)in_sizes; (void)n_in; (void)out_size; (void)ws_size;
  const float* context = (const float*)d_in[0];   // [B, N, D] f32
  const float* W       = (const float*)d_in[1];   // [H, D]    f32
  const int*   ctxmask = (const int*)d_in[2];     // [B, N]    i32
  float* out = (float*)d_out;                     // [B, N, N] f32
  float* fcbuf = (float*)d_ws;                    // needs B*N*H*4 = 32 MiB

  // 1) fc = relu(context @ W^T)
  k_fc<<<dim3((B_SZ * N_SZ) / 16, (H_SZ / 16) / 4), 128, 0, stream>>>(context, W, fcbuf);
  // 2) fill output with NEG_INF
  k_fill<<<(unsigned)((size_t)B_SZ * N_SZ * N_SZ / 4 / 256), 256, 0, stream>>>((float4*)out);
  // 3) fused att strips + top-k scatter
  k_att<<<B_SZ * (N_SZ / 16), 256, 0, stream>>>(fcbuf, ctxmask, out);
}